// MultiLinear_46875273069380
// MI455X (gfx1250) — compile-verified
//
#include <hip/hip_runtime.h>
#include <hip/hip_bf16.h>

typedef __attribute__((ext_vector_type(2))) float v2f;
typedef __attribute__((ext_vector_type(8))) float v8f;

#define IN_DIM  128
#define OUT_DIM 128
#define OPT     64
#define TILE_M  16
#define LDS_STRIDE 132   // 128 + 4 pad -> bank = (4*m + k) % 64, conflict-free

// ---------------- bucketing kernels ----------------

__global__ void ml_zero_kernel(int* counts, int* cursor) {
    int t = threadIdx.x;
    if (t < OPT) { counts[t] = 0; cursor[t] = 0; }
}

__global__ void ml_count_kernel(const int* __restrict__ idx, int* __restrict__ counts, int B) {
    int i = blockIdx.x * blockDim.x + threadIdx.x;
    if (i < B) atomicAdd(&counts[idx[i] & (OPT - 1)], 1);
}

__global__ void ml_scan_kernel(const int* __restrict__ counts, int* __restrict__ offsets) {
    if (threadIdx.x == 0 && blockIdx.x == 0) {
        int acc = 0;
        offsets[0] = 0;
        for (int o = 0; o < OPT; ++o) { acc += counts[o]; offsets[o + 1] = acc; }
    }
}

__global__ void ml_scatter_kernel(const int* __restrict__ idx,
                                  const int* __restrict__ offsets,
                                  int* __restrict__ cursor,
                                  int* __restrict__ perm, int B) {
    int i = blockIdx.x * blockDim.x + threadIdx.x;
    if (i < B) {
        int o = idx[i] & (OPT - 1);
        int pos = atomicAdd(&cursor[o], 1);
        perm[offsets[o] + pos] = i;
    }
}

// ---------------- grouped-GEMM WMMA kernel ----------------
// grid = (OPT, NSLICE); block = 128 threads = 4 waves.
// Block handles 16-row tiles of expert blockIdx.x, strided by gridDim.y.
// Each wave computes a 16x32 slab of the 16x128 output tile with four
// v_wmma_f32_16x16x4_f32 accumulator chains (even/odd k-groups) for ILP.

__global__ void __launch_bounds__(128)
ml_gemm_kernel(const float* __restrict__ inputs,
               const float* __restrict__ w,
               const float* __restrict__ bias,
               const int* __restrict__ offsets,
               const int* __restrict__ perm,
               float* __restrict__ out) {
    __shared__ float As[TILE_M * LDS_STRIDE];
    __shared__ int   rowsS[TILE_M];

    const int o   = blockIdx.x;
    const int tid = threadIdx.x;
    const int beg = offsets[o];
    const int end = offsets[o + 1];
    const int cnt = end - beg;
    if (cnt == 0) return;
    const int nt = (cnt + TILE_M - 1) / TILE_M;

    const float* wb = w + (size_t)o * IN_DIM * OUT_DIM;

    // Pull this expert's 64KB weight matrix toward the WGP caches.
    // 128 threads x 4 cachelines (128B) covers 64KB.
    #pragma unroll
    for (int p = 0; p < 4; ++p)
        __builtin_prefetch(wb + (tid * 4 + p) * 32, 0, 3);

    const int wave = tid >> 5;
    const int lane = tid & 31;
    const int n0   = wave * 32;
    const int m    = lane & 15;          // row within fragment
    const int kh   = (lane >> 4) * 2;    // K sub-offset per ISA A/B layout
    const int n    = n0 + m;             // output column (low half)

    const float bn0 = bias[o * OUT_DIM + n];
    const float bn1 = bias[o * OUT_DIM + n + 16];

    for (int t = blockIdx.y; t < nt; t += gridDim.y) {
        __syncthreads();                 // previous tile fully consumed
        if (tid < TILE_M) {
            int r = beg + t * TILE_M + tid;
            rowsS[tid] = (r < end) ? perm[r] : -1;
        }
        __syncthreads();

        // Gather 16 input rows (2048 floats) straight into LDS with the
        // CDNA5 async global->LDS path (ASYNCcnt). Invalid rows are clamped
        // to the tile's first (always valid) row; their results are never
        // stored, so no zero-fill branch is needed and all lanes issue.
        const int row0 = rowsS[0];
        #pragma unroll
        for (int it = 0; it < 4; ++it) {
            int lin = tid + 128 * it;            // 0..511 float4 slots
            int row = lin >> 5;                  // 32 float4 per row
            int c4  = (lin & 31) * 4;
            int gr  = rowsS[row];
            if (gr < 0) gr = row0;
            unsigned     ldsa = (unsigned)(uintptr_t)(As + row * LDS_STRIDE + c4);
            const float* gp   = inputs + (size_t)gr * IN_DIM + c4;
            asm volatile("global_load_async_to_lds_b128 %0, %1, off"
                         :: "v"(ldsa), "v"(gp) : "memory");
        }
        asm volatile("s_wait_asynccnt 0x0" ::: "memory");
        __syncthreads();

        v8f c0a = {}, c0b = {};
        v8f c1a = {}, c1b = {};
        #pragma unroll 2
        for (int k0 = 0; k0 < IN_DIM; k0 += 8) {
            // ---- even k-group -> chain a ----
            {
                const int kb = k0 + kh;
                v2f a = *(const v2f*)(As + m * LDS_STRIDE + kb);
                v2f b0, b1;
                b0.x = wb[(size_t)kb * OUT_DIM + n];
                b0.y = wb[(size_t)(kb + 1) * OUT_DIM + n];
                b1.x = wb[(size_t)kb * OUT_DIM + n + 16];
                b1.y = wb[(size_t)(kb + 1) * OUT_DIM + n + 16];
                c0a = __builtin_amdgcn_wmma_f32_16x16x4_f32(false, a, false, b0,
                                                            (short)0, c0a, false, false);
                c1a = __builtin_amdgcn_wmma_f32_16x16x4_f32(false, a, false, b1,
                                                            (short)0, c1a, false, false);
            }
            // ---- odd k-group -> chain b ----
            {
                const int kb = k0 + 4 + kh;
                v2f a = *(const v2f*)(As + m * LDS_STRIDE + kb);
                v2f b0, b1;
                b0.x = wb[(size_t)kb * OUT_DIM + n];
                b0.y = wb[(size_t)(kb + 1) * OUT_DIM + n];
                b1.x = wb[(size_t)kb * OUT_DIM + n + 16];
                b1.y = wb[(size_t)(kb + 1) * OUT_DIM + n + 16];
                c0b = __builtin_amdgcn_wmma_f32_16x16x4_f32(false, a, false, b0,
                                                            (short)0, c0b, false, false);
                c1b = __builtin_amdgcn_wmma_f32_16x16x4_f32(false, a, false, b1,
                                                            (short)0, c1b, false, false);
            }
        }

        // D layout: VGPR v -> row (v + 8*(lane>>4)), col = n / n+16.
        // Hoist the 8 row ids into two 128-bit LDS loads, then store.
        const int mbase = (lane >> 4) * 8;
        const int4 ra = *(const int4*)&rowsS[mbase];
        const int4 rb = *(const int4*)&rowsS[mbase + 4];
        int rows8[8] = { ra.x, ra.y, ra.z, ra.w, rb.x, rb.y, rb.z, rb.w };
        #pragma unroll
        for (int v = 0; v < 8; ++v) {
            int gr = rows8[v];
            if (gr >= 0) {
                out[(size_t)gr * OUT_DIM + n]      = (c0a[v] + c0b[v]) + bn0;
                out[(size_t)gr * OUT_DIM + n + 16] = (c1a[v] + c1b[v]) + bn1;
            }
        }
    }
}

// ---------------- launch ----------------

extern "C" void kernel_launch(void* const* d_in, const int* in_sizes, int n_in,
                              void* d_out, int out_size, void* d_ws, size_t ws_size,
                              hipStream_t stream) {
    const float* inputs = (const float*)d_in[0];   // [B, 128]
    const float* w      = (const float*)d_in[1];   // [64, 128, 128]
    const float* bias   = (const float*)d_in[2];   // [64, 128]
    const int*   idx    = (const int*)d_in[3];     // [B]
    float* out          = (float*)d_out;           // [B, 128]

    const int B = in_sizes[0] / IN_DIM;

    // workspace layout (ints)
    int* counts  = (int*)d_ws;             // [64]
    int* cursor  = counts + OPT;           // [64]
    int* offsets = cursor + OPT;           // [65]
    int* perm    = offsets + OPT + 2;      // [B]

    ml_zero_kernel<<<1, OPT, 0, stream>>>(counts, cursor);
    ml_count_kernel<<<(B + 255) / 256, 256, 0, stream>>>(idx, counts, B);
    ml_scan_kernel<<<1, 1, 0, stream>>>(counts, offsets);
    ml_scatter_kernel<<<(B + 255) / 256, 256, 0, stream>>>(idx, offsets, cursor, perm, B);

    dim3 grid(OPT, 16);
    ml_gemm_kernel<<<grid, 128, 0, stream>>>(inputs, w, bias, offsets, perm, out);
}